// Q_shuffle2_52029233823803
// MI455X (gfx1250) — compile-verified
//
#include <hip/hip_runtime.h>

typedef __attribute__((ext_vector_type(2))) float v2f;
typedef __attribute__((ext_vector_type(8))) float v8f;

// D = A(16x4 f32) * B(4x16 f32) + C(16x16 f32), wave32.
// 8-arg form per probe: (neg_a, A, neg_b, B, c_mod, C, reuse_a, reuse_b)
#define WMMA_F32(a, b, c) \
  __builtin_amdgcn_wmma_f32_16x16x4_f32(false, (a), false, (b), (short)0, (c), false, false)

__device__ __forceinline__ v8f zero8() {
  v8f z = {0.f, 0.f, 0.f, 0.f, 0.f, 0.f, 0.f, 0.f};
  return z;
}

// ---------------------------------------------------------------------------
// GEMM: C[8192 x Ncols] = A[8192 x 768] @ Bw[768 x Ncols], fp32 WMMA 16x16x4.
// Block = 128 threads (4 waves). Tile 64(M) x 64(N), K-chunk 16.
// mode 0: plain store to outp. mode 1: scatter into Q/K/V [b,h,n,d] layouts.
// ---------------------------------------------------------------------------
__global__ __launch_bounds__(128) void gemm768_wmma(
    const float* __restrict__ A, const float* __restrict__ Bw,
    int Ncols, int mode, float* __restrict__ outp,
    float* __restrict__ qo, float* __restrict__ ko, float* __restrict__ vo) {
  __shared__ __align__(16) float As[64 * 20];  // 64 rows, stride 20 (bank-safe, 16B-aligned rows)
  __shared__ __align__(16) float Bs[16 * 64];  // 16 rows, stride 64

  const int tid = threadIdx.x;
  const int lane = tid & 31;
  const int wave = tid >> 5;
  const int ml = lane & 15;
  const int hi = lane >> 4;
  const int rowBase = blockIdx.y * 64;
  const int colBase = blockIdx.x * 64;
  const int m0 = wave * 16;  // 16-row strip per wave

  v8f acc[4];
#pragma unroll
  for (int i = 0; i < 4; ++i) acc[i] = zero8();

  for (int k0 = 0; k0 < 768; k0 += 16) {
    __syncthreads();
    // Stage A tile: 64x16 = 256 float4
#pragma unroll
    for (int i = 0; i < 2; ++i) {
      int idx = tid + i * 128;               // 0..255
      int r = idx >> 2, c4 = (idx & 3) << 2;
      float4 vv = *(const float4*)&A[(size_t)(rowBase + r) * 768 + k0 + c4];
      *(float4*)&As[r * 20 + c4] = vv;       // byte offset 80r+16k -> 16B aligned
    }
    // Stage B tile: 16x64 = 256 float4
#pragma unroll
    for (int i = 0; i < 2; ++i) {
      int idx = tid + i * 128;
      int r = idx >> 4, c4 = (idx & 15) << 2;
      float4 vv = *(const float4*)&Bw[(size_t)(k0 + r) * Ncols + colBase + c4];
      *(float4*)&Bs[r * 64 + c4] = vv;
    }
    __syncthreads();

#pragma unroll
    for (int kk = 0; kk < 16; kk += 4) {
      v2f a = *(const v2f*)&As[(m0 + ml) * 20 + kk + 2 * hi];  // A[m][2hi], A[m][2hi+1]
#pragma unroll
      for (int ni = 0; ni < 4; ++ni) {
        v2f b;
        b.x = Bs[(kk + 2 * hi) * 64 + ni * 16 + ml];           // B[2hi][n]
        b.y = Bs[(kk + 2 * hi + 1) * 64 + ni * 16 + ml];       // B[2hi+1][n]
        acc[ni] = WMMA_F32(a, b, acc[ni]);
      }
    }
  }

  // Epilogue: D[v + 8*hi][lane&15] per C/D layout.
#pragma unroll
  for (int ni = 0; ni < 4; ++ni) {
#pragma unroll
    for (int v = 0; v < 8; ++v) {
      int r = rowBase + m0 + v + 8 * hi;
      int c = colBase + ni * 16 + ml;
      float val = acc[ni][v];
      if (mode == 0) {
        outp[(size_t)r * Ncols + c] = val;
      } else {
        int sec = c / 768;            // 0=Q, 1=K, 2=V
        int cc = c % 768;
        int h = cc >> 6, d = cc & 63; // head, dh
        int bb = r >> 10, n = r & 1023;
        size_t dst = ((((size_t)bb * 12 + h) << 10) + n) * 64 + d;
        float* p = (sec == 0) ? qo : (sec == 1) ? ko : vo;
        p[dst] = val;
      }
    }
  }
}

// ---------------------------------------------------------------------------
// Q gather: shuffled_q[bh][i] = qraw[bh][perm[i]], i over n*dh = 65536.
// ---------------------------------------------------------------------------
__global__ __launch_bounds__(256) void qshuffle_kernel(
    const float* __restrict__ qraw, const int* __restrict__ perm,
    float* __restrict__ qs) {
  int g = blockIdx.x * 256 + threadIdx.x;  // 96 * 65536 total
  int i = g & 65535;
  size_t base = (size_t)(g >> 16) << 16;
  qs[base + i] = qraw[base + (size_t)perm[i]];
}

// ---------------------------------------------------------------------------
// Flash attention per (b,h): Q[1024x64] K[1024x64] V[1024x64] -> ctx[1024x64].
// Block = 128 threads (4 waves) owns 64 query rows; loops over 16 key tiles.
// Online softmax; WMMA f32 for both S = Q K^T and O += P V.
// Output written to [B, N, INNER] so the out-projection reads it row-major.
// ---------------------------------------------------------------------------
__global__ __launch_bounds__(128) void flash_attn_wmma(
    const float* __restrict__ qsg, const float* __restrict__ kb,
    const float* __restrict__ vb, float* __restrict__ outp) {
  __shared__ __align__(16) float Ks[64 * 68];
  __shared__ __align__(16) float Vs[64 * 68];
  __shared__ __align__(16) float Qs[64 * 68];  // Q staging; reused as per-wave P buffer

  const int tid = threadIdx.x;
  const int lane = tid & 31;
  const int wave = tid >> 5;
  const int ml = lane & 15;
  const int hi = lane >> 4;
  const int qt = blockIdx.x;  // query tile 0..15
  const int bh = blockIdx.y;  // 0..95
  const size_t base = (size_t)bh << 16;  // 1024*64 elems per (b,h)
  const float* Qg = qsg + base + qt * 4096;
  const float* Kg = kb + base;
  const float* Vg = vb + base;
  const int m0 = wave * 16;

  // Stage Q tile (64x64) coalesced, then build pre-scaled A-fragments.
#pragma unroll
  for (int i = 0; i < 8; ++i) {
    int idx = tid + i * 128;                 // 0..1023
    int r = idx >> 4, c4 = (idx & 15) << 2;
    *(float4*)&Qs[r * 68 + c4] = *(const float4*)&Qg[r * 64 + c4];
  }
  __syncthreads();

  v2f qa[16];
#pragma unroll
  for (int t = 0; t < 16; ++t) {
    v2f a = *(const v2f*)&Qs[(m0 + ml) * 68 + 4 * t + 2 * hi];
    qa[t] = a * 0.125f;  // 1/sqrt(64)
  }

  v8f o[4];
  float rmax[8], rsum[8];
#pragma unroll
  for (int i = 0; i < 4; ++i) o[i] = zero8();
#pragma unroll
  for (int v = 0; v < 8; ++v) { rmax[v] = -1e30f; rsum[v] = 0.f; }

  for (int jt = 0; jt < 16; ++jt) {
    __syncthreads();  // K/V (and P region) free to overwrite
    const float* Kj = Kg + jt * 4096;
    const float* Vj = Vg + jt * 4096;
#pragma unroll
    for (int i = 0; i < 8; ++i) {
      int idx = tid + i * 128;
      int r = idx >> 4, c4 = (idx & 15) << 2;
      *(float4*)&Ks[r * 68 + c4] = *(const float4*)&Kj[r * 64 + c4];
      *(float4*)&Vs[r * 68 + c4] = *(const float4*)&Vj[r * 64 + c4];
    }
    __syncthreads();

    // S = Q K^T : B[k][n] = K[key=n][feat=k]
    v8f s[4];
#pragma unroll
    for (int i = 0; i < 4; ++i) s[i] = zero8();
#pragma unroll
    for (int t = 0; t < 16; ++t) {
      const int kk = 4 * t;
      v2f a = qa[t];
#pragma unroll
      for (int ni = 0; ni < 4; ++ni) {
        v2f b = *(const v2f*)&Ks[(ni * 16 + ml) * 68 + kk + 2 * hi];
        s[ni] = WMMA_F32(a, b, s[ni]);
      }
    }

    // Online softmax. Row (v + 8*hi) lives across 16 lanes of this half-wave.
#pragma unroll
    for (int v = 0; v < 8; ++v) {
      float mx = fmaxf(fmaxf(s[0][v], s[1][v]), fmaxf(s[2][v], s[3][v]));
      mx = fmaxf(mx, __shfl_xor(mx, 1, 32));
      mx = fmaxf(mx, __shfl_xor(mx, 2, 32));
      mx = fmaxf(mx, __shfl_xor(mx, 4, 32));
      mx = fmaxf(mx, __shfl_xor(mx, 8, 32));
      float mnew = fmaxf(rmax[v], mx);
      float corr = __expf(rmax[v] - mnew);
      rmax[v] = mnew;
      float ps = 0.f;
#pragma unroll
      for (int ni = 0; ni < 4; ++ni) {
        float p = __expf(s[ni][v] - mnew);
        s[ni][v] = p;
        ps += p;
        o[ni][v] *= corr;
      }
      ps += __shfl_xor(ps, 1, 32);
      ps += __shfl_xor(ps, 2, 32);
      ps += __shfl_xor(ps, 4, 32);
      ps += __shfl_xor(ps, 8, 32);
      rsum[v] = rsum[v] * corr + ps;
    }

    // C-layout -> A-layout via per-wave LDS region (rows m0..m0+15 of Qs).
    float* Pw = &Qs[m0 * 68];
#pragma unroll
    for (int ni = 0; ni < 4; ++ni) {
#pragma unroll
      for (int v = 0; v < 8; ++v) {
        Pw[(v + 8 * hi) * 68 + ni * 16 + ml] = s[ni][v];
      }
    }
    __syncthreads();

    // O += P @ V : A = P[16 x 64], B[k][n] = V[key=k][dh=n]
#pragma unroll
    for (int t = 0; t < 16; ++t) {
      const int kk = 4 * t;
      v2f a = *(const v2f*)&Pw[ml * 68 + kk + 2 * hi];
#pragma unroll
      for (int ni = 0; ni < 4; ++ni) {
        v2f b;
        b.x = Vs[(kk + 2 * hi) * 68 + ni * 16 + ml];
        b.y = Vs[(kk + 2 * hi + 1) * 68 + ni * 16 + ml];
        o[ni] = WMMA_F32(a, b, o[ni]);
      }
    }
  }

  // Normalize and write ctx into [B, N, INNER] (row-major for out-projection).
  const int b = bh / 12, h = bh % 12;
#pragma unroll
  for (int v = 0; v < 8; ++v) {
    float inv = 1.f / rsum[v];
    int n = qt * 64 + m0 + v + 8 * hi;
#pragma unroll
    for (int ni = 0; ni < 4; ++ni) {
      int c = h * 64 + ni * 16 + ml;
      outp[(((size_t)b << 10) + n) * 768 + c] = o[ni][v] * inv;
    }
  }
}

// ---------------------------------------------------------------------------
// Launch: qkv gemm -> q gather -> flash attention -> out projection.
// ---------------------------------------------------------------------------
extern "C" void kernel_launch(void* const* d_in, const int* in_sizes, int n_in,
                              void* d_out, int out_size, void* d_ws, size_t ws_size,
                              hipStream_t stream) {
  (void)in_sizes; (void)n_in; (void)out_size; (void)ws_size;
  const float* x    = (const float*)d_in[0];  // [8,1024,768]
  const float* Wqkv = (const float*)d_in[1];  // [768,2304]
  const float* Wout = (const float*)d_in[2];  // [768,768]
  const int*   perm = (const int*)d_in[3];    // [65536]
  float* out = (float*)d_out;                 // [8,1024,768]

  float* ws = (float*)d_ws;
  const size_t QSZ = 6291456;  // B*H*N*DH
  float* qraw = ws;            // dead after gather
  float* kbuf = ws + QSZ;
  float* vbuf = ws + 2 * QSZ;
  float* qshf = ws + 3 * QSZ;
  float* attn = ws;            // aliases qraw (context, [B,N,768])

  // 1) QKV projection: [8192x768] @ [768x2304], scatter into q/k/v layouts
  gemm768_wmma<<<dim3(36, 128), 128, 0, stream>>>(x, Wqkv, 2304, 1, nullptr,
                                                  qraw, kbuf, vbuf);
  // 2) Fixed-permutation gather on flattened Q per (b,h)
  qshuffle_kernel<<<24576, 256, 0, stream>>>(qraw, perm, qshf);
  // 3) Flash attention over 96 (b,h) slabs x 16 query tiles
  flash_attn_wmma<<<dim3(16, 96), 128, 0, stream>>>(qshf, kbuf, vbuf, attn);
  // 4) Output projection: [8192x768] @ [768x768] -> d_out
  gemm768_wmma<<<dim3(12, 128), 128, 0, stream>>>(attn, Wout, 768, 0, out,
                                                  nullptr, nullptr, nullptr);
}